// SPNPoolingConv_55697135895301
// MI455X (gfx1250) — compile-verified
//
#include <hip/hip_runtime.h>
#include <math.h>

// SPN ROI pooling (bilinear crop-resize + sigmoid gate) for MI455X (gfx1250).
//
// Shapes (compile-time constants from the reference):
//   fmap   : (B=2, T=16, H=64, W=64, C=256) f32   -> 128 MB (fits 192 MB L2)
//   spn_cls: (B, R=32) f32
//   spn_reg: (B, R, 2) f32
//   out    : (B, T, R, P=7, P=7, C=256) f32       -> 51.4 MB, write-once
//
// Memory-bound gather (AI ~0.5 FLOP/B): no matmul structure -> no WMMA.
// Strategy:
//  - one block per (b, r, py, px) via 3D grid (no div/rem at block start)
//  - 4 bilinear tap rows are contiguous 1KB channel vectors -> 64 lanes x
//    float4 = fully coalesced global_load_b128 in wave32
//  - 256 threads (8 wave32) cover 4 frames per pass, 4 unrolled passes (T=16)
//  - global_prefetch_b8 (gfx1250) prefetches the next frame-slab's tap rows
//  - non-temporal b128 stores keep the 192MB L2 dedicated to the re-read fmap

#define B_   2
#define T_   16
#define H_   64
#define W_   64
#define C_   256
#define R_   32
#define P_   7
#define DEV_ 0.1f

// Native clang vector type: required by __builtin_nontemporal_store, and
// loads/stores of it lower to global_{load,store}_b128.
typedef float v4f __attribute__((ext_vector_type(4)));

__device__ __forceinline__ float sigmoidf_(float x) {
    return 1.0f / (1.0f + __expf(-x));
}

__global__ __launch_bounds__(256) void spn_roi_pool_kernel(
    const float* __restrict__ fmap,
    const float* __restrict__ spn_cls,
    const float* __restrict__ spn_reg,
    float* __restrict__ out)
{
    // 3D grid: x = (py,px) in [0,49), y = r, z = b  -> no integer div/rem
    const int pp = blockIdx.x;
    const int r  = blockIdx.y;
    const int b  = blockIdx.z;
    const int py = pp / P_;     // constant-divisor (7): strength-reduced
    const int px = pp % P_;

    // ---- per-ROI tap computation (uniform across the block; cheap) ----
    const float cy = sigmoidf_(spn_reg[(b * R_ + r) * 2 + 0]);
    const float cx = sigmoidf_(spn_reg[(b * R_ + r) * 2 + 1]);

    const float y1 = fminf(fmaxf(cy - DEV_, 0.0f), 1.0f);
    const float y2 = fminf(fmaxf(cy + DEV_, 0.0f), 1.0f);
    const float x1 = fminf(fmaxf(cx - DEV_, 0.0f), 1.0f);
    const float x2 = fminf(fmaxf(cx + DEV_, 0.0f), 1.0f);

    const float ty = (float)py * (1.0f / (float)(P_ - 1));  // linspace(0,1,P)[py]
    const float tx = (float)px * (1.0f / (float)(P_ - 1));

    const float ysf = (y1 + ty * (y2 - y1)) * (float)(H_ - 1);
    const float xsf = (x1 + tx * (x2 - x1)) * (float)(W_ - 1);

    const int y0  = min(max((int)floorf(ysf), 0), H_ - 1);
    const int x0  = min(max((int)floorf(xsf), 0), W_ - 1);
    const int y1i = min(y0 + 1, H_ - 1);
    const int x1i = min(x0 + 1, W_ - 1);

    const float wy = ysf - (float)y0;
    const float wx = xsf - (float)x0;

    const float gate = sigmoidf_(spn_cls[b * R_ + r]);

    // Fold gate into the 4 bilinear weights.
    const float w00 = gate * (1.0f - wy) * (1.0f - wx);
    const float w01 = gate * (1.0f - wy) * wx;
    const float w10 = gate * wy * (1.0f - wx);
    const float w11 = gate * wy * wx;

    // ---- data movement: 64 lanes x float4 = one 1KB channel row ----
    const int c4 = threadIdx.x & 63;   // float4 group within the channel row
    const int tq = threadIdx.x >> 6;   // which of 4 frames in this pass

    const size_t tstride = (size_t)H_ * W_ * C_;              // per-frame stride
    const size_t bbase   = (size_t)b * T_ * tstride;

    const size_t o00 = (size_t)(y0  * W_ + x0 ) * C_;
    const size_t o01 = (size_t)(y0  * W_ + x1i) * C_;
    const size_t o10 = (size_t)(y1i * W_ + x0 ) * C_;
    const size_t o11 = (size_t)(y1i * W_ + x1i) * C_;

    const size_t coff = (size_t)c4 * 4;   // float offset of this lane's float4

#pragma unroll
    for (int i = 0; i < 4; ++i) {
        const int t = i * 4 + tq;
        const float* base = fmap + bbase + (size_t)t * tstride;

        // Prefetch the next frame-slab's tap rows (gfx1250 global_prefetch_b8)
        // while this slab's loads are in flight.
        if (i < 3) {
            const float* nbase = base + 4 * tstride;
            __builtin_prefetch(nbase + o00 + coff, 0, 3);
            __builtin_prefetch(nbase + o01 + coff, 0, 3);
            __builtin_prefetch(nbase + o10 + coff, 0, 3);
            __builtin_prefetch(nbase + o11 + coff, 0, 3);
        }

        const v4f v00 = *(const v4f*)(base + o00 + coff);
        const v4f v01 = *(const v4f*)(base + o01 + coff);
        const v4f v10 = *(const v4f*)(base + o10 + coff);
        const v4f v11 = *(const v4f*)(base + o11 + coff);

        v4f acc = w00 * v00 + w01 * v01 + w10 * v10 + w11 * v11;

        // out[b][t][r][py][px][c] — write-once stream: non-temporal store so
        // the 192MB L2 stays dedicated to the re-read fmap working set.
        const size_t oidx =
            (((((size_t)(b * T_ + t) * R_ + r) * P_ + py) * P_ + px) * C_) + coff;
        __builtin_nontemporal_store(acc, (v4f*)(out + oidx));
    }
}

extern "C" void kernel_launch(void* const* d_in, const int* in_sizes, int n_in,
                              void* d_out, int out_size, void* d_ws, size_t ws_size,
                              hipStream_t stream) {
    const float* fmap    = (const float*)d_in[0];
    const float* spn_cls = (const float*)d_in[1];
    const float* spn_reg = (const float*)d_in[2];
    float* out = (float*)d_out;

    // 3136 blocks x 256 threads (8 wave32 each)
    dim3 grid(P_ * P_, R_, B_);
    spn_roi_pool_kernel<<<grid, dim3(256), 0, stream>>>(
        fmap, spn_cls, spn_reg, out);
}